// GCN_for_emb_20710332301824
// MI455X (gfx1250) — compile-verified
//
#include <hip/hip_runtime.h>
#include <hip/hip_bf16.h>

// ---------------------------------------------------------------------------
// 2-layer GCN (DGL GraphConv, norm='both') for MI455X / gfx1250.
//   agg[dst] += (h * rsqrt(out_deg))[src]      (COO scatter, fp32 HW atomics)
//   out      = relu((agg * rsqrt(in_deg)) @ W + b)   via V_WMMA_F32_16X16X4_F32
// ---------------------------------------------------------------------------

typedef float v2f __attribute__((ext_vector_type(2)));
typedef float v8f __attribute__((ext_vector_type(8)));

#define FEATS 128
#define SMEM_STRIDE 132   // 128 + 4 pad: row-to-row bank rotation in 64-bank LDS

// ---------------- zero fill (float4 grid-stride; n % 4 == 0) ----------------
__global__ void gcn_zero_f32(float* __restrict__ p, size_t n)
{
    size_t n4 = n >> 2;
    size_t i = (size_t)blockIdx.x * blockDim.x + threadIdx.x;
    size_t stride = (size_t)gridDim.x * blockDim.x;
    float4 z = make_float4(0.f, 0.f, 0.f, 0.f);
    for (; i < n4; i += stride)
        ((float4*)p)[i] = z;
}

// ---------------- degree accumulation (one thread per edge) -----------------
__global__ void gcn_deg_accum(const int* __restrict__ src, const int* __restrict__ dst,
                              float* __restrict__ outdeg, float* __restrict__ indeg,
                              int n_edges)
{
    int e = blockIdx.x * blockDim.x + threadIdx.x;
    if (e >= n_edges) return;
    unsafeAtomicAdd(&outdeg[src[e]], 1.0f);   // global_atomic_add_f32
    unsafeAtomicAdd(&indeg[dst[e]], 1.0f);
}

// ---------------- deg -> rsqrt(max(deg,1)) in place -------------------------
__global__ void gcn_deg_rsqrt(float* __restrict__ d, int n)
{
    int i = blockIdx.x * blockDim.x + threadIdx.x;
    if (i >= n) return;
    float v = d[i];
    v = v < 1.0f ? 1.0f : v;
    d[i] = rsqrtf(v);
}

// ---------------- row scaling: out[i,:] = in[i,:] * rs[i] -------------------
__global__ void gcn_scale_rows(const float* __restrict__ in, const float* __restrict__ rs,
                               float* __restrict__ out, int n_nodes)
{
    size_t idx = (size_t)blockIdx.x * blockDim.x + threadIdx.x; // one float4 each
    size_t total = (size_t)n_nodes * (FEATS / 4);
    if (idx >= total) return;
    int node = (int)(idx >> 5);               // 32 float4 per row
    float r = rs[node];
    float4 v = ((const float4*)in)[idx];
    v.x *= r; v.y *= r; v.z *= r; v.w *= r;
    ((float4*)out)[idx] = v;
}

// ---------------- edge scatter-add: one wave32 per edge ---------------------
// Each lane gathers a float4 of the source row and issues 4 fp32 HW atomics.
// Node rows (51 MB) are L2-resident on MI455X (192 MB L2).
__global__ __launch_bounds__(256) void gcn_scatter_edges(
    const int* __restrict__ src, const int* __restrict__ dst,
    const float* __restrict__ hs, float* __restrict__ agg, int n_edges)
{
    int wave = (int)(((size_t)blockIdx.x * blockDim.x + threadIdx.x) >> 5);
    int lane = threadIdx.x & 31;
    if (wave >= n_edges) return;
    int s = src[wave];
    int d = dst[wave];
    float4 v = ((const float4*)(hs + (size_t)s * FEATS))[lane];
    float* dp = agg + (size_t)d * FEATS + lane * 4;
    unsafeAtomicAdd(dp + 0, v.x);
    unsafeAtomicAdd(dp + 1, v.y);
    unsafeAtomicAdd(dp + 2, v.z);
    unsafeAtomicAdd(dp + 3, v.w);
}

// ---------------- fused GEMM: out = relu((A*rs_in) @ W + b) -----------------
// Block = 256 threads = 8 wave32; block owns a 16-row strip (n_nodes % 16==0),
// wave w computes the 16x16 tile of columns [16w, 16w+16) via 32 chained
// v_wmma_f32_16x16x4_f32. A-strip is staged once in LDS, pre-scaled by rs_in.
__global__ __launch_bounds__(256) void gcn_gemm_wmma(
    const float* __restrict__ A,      // [n,128] aggregated (unnormalized dst side)
    const float* __restrict__ rs_in,  // [n]
    const float* __restrict__ W,      // [128,128] row-major (k, j)
    const float* __restrict__ bias,   // [128]
    float* __restrict__ out,          // [n,128]
    int n_nodes)
{
    __shared__ float sA[16 * SMEM_STRIDE];

    const int tid  = threadIdx.x;
    const int row0 = blockIdx.x * 16;

    // Stage A tile into LDS, scaled by rsqrt(in_deg). 256 thr x 8 floats.
    {
        int r = tid >> 4;              // 0..15
        int c = (tid & 15) * 8;        // 0..120
        int grow = row0 + r;
        float rs = rs_in[grow];
        const float4* gp = (const float4*)(A + (size_t)grow * FEATS + c);
        float4 v0 = gp[0];
        float4 v1 = gp[1];
        v0.x *= rs; v0.y *= rs; v0.z *= rs; v0.w *= rs;
        v1.x *= rs; v1.y *= rs; v1.z *= rs; v1.w *= rs;
        float4* sp = (float4*)(sA + r * SMEM_STRIDE + c); // 528B row stride: 16B aligned
        sp[0] = v0;
        sp[1] = v1;
    }
    __syncthreads();

    const int wave = tid >> 5;         // 0..7 -> column tile
    const int lane = tid & 31;
    const int half = lane >> 4;        // 0 or 1
    const int l16  = lane & 15;
    const int col  = wave * 16 + l16;  // global output column for B/C/D lanes

    const float* sArow = sA + l16 * SMEM_STRIDE;

    v8f acc = {0.f, 0.f, 0.f, 0.f, 0.f, 0.f, 0.f, 0.f};

    // A 16x4 fp32 layout: lane<16 -> K = kb+{0,1}; lane>=16 -> K = kb+{2,3}
    // B 4x16 fp32 layout: vgpr0 = row K(=kb+2*half), vgpr1 = row K+1, N = l16
    #pragma unroll
    for (int kb = 0; kb < FEATS; kb += 4) {
        int k0 = kb + half * 2;
        v2f a;
        a.x = sArow[k0];               // ds_load_b64
        a.y = sArow[k0 + 1];
        v2f b;
        b.x = W[(size_t)k0 * FEATS + col];
        b.y = W[(size_t)(k0 + 1) * FEATS + col];
        acc = __builtin_amdgcn_wmma_f32_16x16x4_f32(
            /*neg_a=*/false, a, /*neg_b=*/false, b,
            /*c_mod=*/(short)0, acc, /*reuse_a=*/false, /*reuse_b=*/false);
    }

    // C/D 16x16 fp32 layout: vgpr r, lanes0-15 -> M=r, lanes16-31 -> M=r+8
    float bb = bias[col];
    #pragma unroll
    for (int r = 0; r < 8; ++r) {
        int grow = row0 + r + half * 8;
        float v = acc[r] + bb;
        out[(size_t)grow * FEATS + col] = v > 0.f ? v : 0.f;
    }
}

// ---------------------------------------------------------------------------
extern "C" void kernel_launch(void* const* d_in, const int* in_sizes, int n_in,
                              void* d_out, int out_size, void* d_ws, size_t ws_size,
                              hipStream_t stream)
{
    const float* features = (const float*)d_in[0];
    const float* W1       = (const float*)d_in[1];
    const float* b1       = (const float*)d_in[2];
    const float* W2       = (const float*)d_in[3];
    const float* b2       = (const float*)d_in[4];
    const int*   src      = (const int*)d_in[5];
    const int*   dst      = (const int*)d_in[6];
    float*       out      = (float*)d_out;

    const int n_nodes = in_sizes[0] / FEATS;   // 100000 (multiple of 16)
    const int n_edges = in_sizes[5];           // 1600000

    // Workspace: rs_out[N] | rs_in[N] | bufA[N*128] | bufB[N*128]  (~103 MB)
    float* ws     = (float*)d_ws;
    size_t degPad = ((size_t)2 * n_nodes + 3) & ~(size_t)3;
    float* rs_out = ws;
    float* rs_in  = ws + n_nodes;
    float* bufA   = ws + degPad;
    float* bufB   = bufA + (size_t)n_nodes * FEATS;

    const int T = 256;
    const size_t featElems = (size_t)n_nodes * FEATS;
    const int zeroBlocksDeg  = 256;
    const int zeroBlocksFeat = 4096;
    const int degBlocks   = (n_edges + T - 1) / T;
    const int rsqBlocks   = (2 * n_nodes + T - 1) / T;
    const int scaleBlocks = (int)(((size_t)n_nodes * (FEATS / 4) + T - 1) / T);
    const int scatBlocks  = (n_edges + (T / 32) - 1) / (T / 32);  // 8 edges/block
    const int gemmBlocks  = n_nodes / 16;

    // ---- degrees -> rsqrt(max(deg,1)) ----
    gcn_zero_f32<<<zeroBlocksDeg, T, 0, stream>>>(rs_out, degPad);
    gcn_deg_accum<<<degBlocks, T, 0, stream>>>(src, dst, rs_out, rs_in, n_edges);
    gcn_deg_rsqrt<<<rsqBlocks, T, 0, stream>>>(rs_out, 2 * n_nodes);

    // ---- layer 1 ----
    gcn_scale_rows<<<scaleBlocks, T, 0, stream>>>(features, rs_out, bufA, n_nodes);
    gcn_zero_f32<<<zeroBlocksFeat, T, 0, stream>>>(bufB, featElems);
    gcn_scatter_edges<<<scatBlocks, T, 0, stream>>>(src, dst, bufA, bufB, n_edges);
    gcn_gemm_wmma<<<gemmBlocks, T, 0, stream>>>(bufB, rs_in, W1, b1, bufA, n_nodes);

    // ---- layer 2 ----
    gcn_scale_rows<<<scaleBlocks, T, 0, stream>>>(bufA, rs_out, bufA, n_nodes);
    gcn_zero_f32<<<zeroBlocksFeat, T, 0, stream>>>(bufB, featElems);
    gcn_scatter_edges<<<scatBlocks, T, 0, stream>>>(src, dst, bufA, bufB, n_edges);
    gcn_gemm_wmma<<<gemmBlocks, T, 0, stream>>>(bufB, rs_in, W2, b2, out, n_nodes);
}